// MinEuclideanDistBlock_28321014350352
// MI455X (gfx1250) — compile-verified
//
#include <hip/hip_runtime.h>
#include <math.h>

#define Bsz 64
#define Cch 3
#define Lx  2048
#define Sx  64
#define Nx  256
#define Wx  (Lx - Sx + 1)   // 1985
#define WT  64               // w rows per block
#define XSL (WT + Sx)        // 128: x slice floats per channel per block
#define NWT ((Wx + WT - 1) / WT)  // 32

typedef __attribute__((ext_vector_type(2))) float v2f;
typedef __attribute__((ext_vector_type(8))) float v8f;

// ---------------------------------------------------------------------------
// Prelude: shp_sq[c][n] = sum_s shapelets[c][n][s]^2  (into workspace)
//          out_bits[b*N+n] = +inf bit pattern (monotone-min accumulator)
// ---------------------------------------------------------------------------
__global__ __launch_bounds__(256) void shapelet_prelude(
    const float* __restrict__ shp, float* __restrict__ shp_sq,
    unsigned int* __restrict__ out_bits) {
  int t = blockIdx.x * blockDim.x + threadIdx.x;
  if (t < Cch * Nx) {
    const float* p = shp + (size_t)t * Sx;
    float acc = 0.f;
#pragma unroll 8
    for (int s = 0; s < Sx; ++s) { float v = p[s]; acc += v * v; }
    shp_sq[t] = acc;
  }
  if (t < Bsz * Nx) out_bits[t] = 0x7F800000u;  // +inf
}

// ---------------------------------------------------------------------------
// Main kernel: one block per (b, 64-row w-tile). 8 wave32; wave j owns
// col-tiles {2j,2j+1} x row-tiles {0..3} = 8 fp32 16x16 accumulators.
// A-side (sliding windows of x) staged once per block into LDS (3x128 f32);
// A fragments then come from ds_load with immediate offsets. B fragments
// from global (L2-resident) with 32-bit offsets. 16 K-steps of
// V_WMMA_F32_16X16X4_F32 per channel; epilogue d = sqrt(max(d2,0)),
// channel-sum, register min over w, shfl half-merge, atomicMin on bits.
// ---------------------------------------------------------------------------
__global__ __launch_bounds__(256) void shapelet_min_dist(
    const float* __restrict__ x, const float* __restrict__ shp,
    const float* __restrict__ shp_sq_g, unsigned int* __restrict__ out_bits) {
  const int b  = blockIdx.x;       // 0..63
  const int w0 = blockIdx.y * WT;  // 0..1984

  const int tid   = threadIdx.x;   // 0..255
  const int wave  = tid >> 5;      // 0..7
  const int lane  = tid & 31;
  const int lhalf = lane >> 4;     // 0/1
  const int l16   = lane & 15;

  __shared__ float s_x[Cch][XSL];      // 1.5 KB: x slice per channel
  __shared__ float s_winsq[Cch][WT];   // 768 B
  __shared__ float s_shpsq[Cch][Nx];   // 3 KB

  // stage x slice (384 floats, coalesced) and ||shp||^2 (768 floats)
  for (int i = tid; i < Cch * XSL; i += 256) {
    const int c = i >> 7;              // XSL == 128
    const int j = i & (XSL - 1);
    int idx = w0 + j; if (idx > Lx - 1) idx = Lx - 1;  // tail clamp (rows masked later)
    s_x[c][j] = x[((size_t)b * Cch + c) * Lx + idx];
  }
  for (int i = tid; i < Cch * Nx; i += 256)
    (&s_shpsq[0][0])[i] = shp_sq_g[i];
  __syncthreads();

  // ||win||^2 for this block's 64 rows, per channel, from the LDS slice
  if (tid < Cch * WT) {
    const int c  = tid >> 6;
    const int wl = tid & (WT - 1);
    float acc = 0.f;
#pragma unroll 8
    for (int s = 0; s < Sx; ++s) { float v = s_x[c][wl + s]; acc += v * v; }
    s_winsq[c][wl] = acc;
  }
  __syncthreads();

  const int nbase0 = 16 * (2 * wave);  // col-tile q=0 base n
  const int nbase1 = nbase0 + 16;      // col-tile q=1 base n

  // Lane-variant parts of fragment addresses (computed once):
  // A (M=l16, K base has +2 for upper lane half), in words within s_x[c]
  const int aoff = l16 + 2 * lhalf;
  // B: shp[c][n][kb], n = nbase + l16, kb = 4k + 2*lhalf, in words
  const unsigned boff0 = (unsigned)(nbase0 + l16) * Sx + 2u * lhalf;
  const unsigned boff1 = (unsigned)(nbase1 + l16) * Sx + 2u * lhalf;

  v8f dsum[4][2];
#pragma unroll
  for (int rt = 0; rt < 4; ++rt)
#pragma unroll
    for (int q = 0; q < 2; ++q) dsum[rt][q] = (v8f)0.f;

  for (int c = 0; c < Cch; ++c) {
    const float* __restrict__ ps = shp + (size_t)c * Nx * Sx;  // shp[c][n][s]

    v8f cross[4][2];
#pragma unroll
    for (int rt = 0; rt < 4; ++rt)
#pragma unroll
      for (int q = 0; q < 2; ++q) cross[rt][q] = (v8f)0.f;

#pragma unroll
    for (int k = 0; k < 16; ++k) {          // K-steps of 4
      // A fragments from LDS: word offset = aoff + 16*rt + 4*k (imm-foldable)
      v2f a[4];
#pragma unroll
      for (int rt = 0; rt < 4; ++rt) {
        const float* pa = &s_x[c][aoff + 16 * rt + 4 * k];
        a[rt].x = pa[0];
        a[rt].y = pa[1];
      }
      // B fragments from global: base + 32-bit offset, k folds to imm offset
      v2f bb[2];
      bb[0].x = ps[boff0 + 4u * k]; bb[0].y = ps[boff0 + 4u * k + 1u];
      bb[1].x = ps[boff1 + 4u * k]; bb[1].y = ps[boff1 + 4u * k + 1u];

#pragma unroll
      for (int rt = 0; rt < 4; ++rt)
#pragma unroll
        for (int q = 0; q < 2; ++q)
          cross[rt][q] = __builtin_amdgcn_wmma_f32_16x16x4_f32(
              false, a[rt], false, bb[q], (short)0, cross[rt][q], false, false);
    }

    // d = sqrt(max(winsq + shpsq - 2*cross, 0)); accumulate across channels
#pragma unroll
    for (int rt = 0; rt < 4; ++rt)
#pragma unroll
      for (int q = 0; q < 2; ++q) {
        const float ssq = s_shpsq[c][(q ? nbase1 : nbase0) + l16];
#pragma unroll
        for (int r = 0; r < 8; ++r) {
          const int wl = 16 * rt + r + 8 * lhalf;   // C/D layout: M = r + 8*lhalf
          const float wsq = s_winsq[c][wl];
          float d2 = wsq + ssq - 2.0f * cross[rt][q][r];
          dsum[rt][q][r] += sqrtf(fmaxf(d2, 0.0f));
        }
      }
  }

  // min over w (mask rows past W), merge lane halves (same n), fold into out
#pragma unroll
  for (int q = 0; q < 2; ++q) {
    float m = __builtin_inff();
#pragma unroll
    for (int rt = 0; rt < 4; ++rt)
#pragma unroll
      for (int r = 0; r < 8; ++r) {
        const int wg = w0 + 16 * rt + r + 8 * lhalf;
        float v = (wg < Wx) ? dsum[rt][q][r] : __builtin_inff();
        m = fminf(m, v);
      }
    m = fminf(m, __shfl_xor(m, 16, 32));  // lanes l and l+16 share n
    if (lane < 16) {
      const int n = (q ? nbase1 : nbase0) + l16;
      atomicMin(&out_bits[b * Nx + n], __float_as_uint(m));
    }
  }
}

// ---------------------------------------------------------------------------
extern "C" void kernel_launch(void* const* d_in, const int* in_sizes, int n_in,
                              void* d_out, int out_size, void* d_ws, size_t ws_size,
                              hipStream_t stream) {
  const float* x   = (const float*)d_in[0];      // (B, C, L) f32
  const float* shp = (const float*)d_in[1];      // (C, N, S) f32
  float* shp_sq          = (float*)d_ws;         // C*N floats
  unsigned int* out_bits = (unsigned int*)d_out; // B*N, min-bits == float value

  shapelet_prelude<<<(Bsz * Nx + 255) / 256, 256, 0, stream>>>(shp, shp_sq, out_bits);

  dim3 grid(Bsz, NWT);  // 64 x 32
  shapelet_min_dist<<<grid, 256, 0, stream>>>(x, shp, shp_sq, out_bits);
}